// ThinPlateSpline_62165356642882
// MI455X (gfx1250) — compile-verified
//
#include <hip/hip_runtime.h>
#include <hip/hip_bf16.h>

typedef float v2f __attribute__((ext_vector_type(2)));
typedef float v8f __attribute__((ext_vector_type(8)));

namespace {
constexpr int kH = 512, kW = 512, kHW = kH * kW;
constexpr int kB = 32, kC = 3, kN = 16;     // batches, channels, ctrl points
constexpr int kMKStride = 19;               // mul_kernel row stride (N+3); cols 16..18 hit zero rows
constexpr float kCanvas = 0.5f;
constexpr int kPitch = 68;                  // LDS row pitch (floats), padded vs 64 to dodge bank conflicts
}

__global__ __launch_bounds__(256, 2) void tps_wmma_sample_kernel(
    const float* __restrict__ inputs,   // (B, C, H, W)
    const float* __restrict__ src,      // (B, 16, 2) ctrl points
    const float* __restrict__ mk,       // (H*W, 19) mul_kernel
    float* __restrict__ out)            // (B, C, H, W)
{
  __shared__ float gridLds[8][16][kPitch];  // [wave][pixel-in-tile][j = 2*b + d]

  const int lane = threadIdx.x & 31;
  const int wave = threadIdx.x >> 5;
  const int tile = blockIdx.x * 8 + wave;   // one 16-pixel tile per wave; HW/16 = 16384 tiles exactly
  const int p0   = tile * 16;
  const int half = lane >> 4;               // 0: lanes 0-15, 1: lanes 16-31
  const int l15  = lane & 15;

  // ---------------- Phase 1: grid GEMM via V_WMMA_F32_16X16X4_F32 ----------------
  // D(16px x 64col) = MK(16x16) * SRC(16x64); column j = 2*b + d (b=batch, d=x/y).
  // B-operand (4x16 per step): lanes hold N, VGPR idx + lane-half hold K (mirror of A layout).
  v2f bmat[4][4];
#pragma unroll
  for (int kc = 0; kc < 4; ++kc) {
    const int ka = kc * 4 + half * 2;       // this lane-half supplies K = ka, ka+1
#pragma unroll
    for (int jt = 0; jt < 4; ++jt) {
      const int j = jt * 16 + l15;
      const int b = j >> 1, d = j & 1;
      v2f bb;
      bb.x = src[(b * kN + ka) * 2 + d];
      bb.y = src[(b * kN + ka + 1) * 2 + d];
      bmat[kc][jt] = bb;
    }
  }

  v8f acc0 = {}, acc1 = {}, acc2 = {}, acc3 = {};
  const long arow = (long)(p0 + l15) * kMKStride;   // A: lane holds row M = l15
#pragma unroll
  for (int kc = 0; kc < 4; ++kc) {
    const int ka = kc * 4 + half * 2;
    v2f a;
    a.x = mk[arow + ka];
    a.y = mk[arow + ka + 1];
    acc0 = __builtin_amdgcn_wmma_f32_16x16x4_f32(false, a, false, bmat[kc][0], (short)0, acc0, false, false);
    acc1 = __builtin_amdgcn_wmma_f32_16x16x4_f32(false, a, false, bmat[kc][1], (short)0, acc1, false, false);
    acc2 = __builtin_amdgcn_wmma_f32_16x16x4_f32(false, a, false, bmat[kc][2], (short)0, acc2, false, false);
    acc3 = __builtin_amdgcn_wmma_f32_16x16x4_f32(false, a, false, bmat[kc][3], (short)0, acc3, false, false);
  }

  // D layout: lane holds column j = nt*16 + l15, rows M = half*8 + m in VGPR m.
#pragma unroll
  for (int m = 0; m < 8; ++m) {
    const int row = half * 8 + m;
    gridLds[wave][row][0 * 16 + l15] = acc0[m];
    gridLds[wave][row][1 * 16 + l15] = acc1[m];
    gridLds[wave][row][2 * 16 + l15] = acc2[m];
    gridLds[wave][row][3 * 16 + l15] = acc3[m];
  }
  __syncthreads();

  // ---------------- Phase 2: bilinear grid_sample + canvas composite ----------------
  // Lanes 0-15 / 16-31 each cover the 16 consecutive pixels of the tile for one batch;
  // 16 iterations x 2 batches covers all 32 batches. Stores are 64B-contiguous per (b,c).
  const int px = l15;
  const int p  = p0 + px;
  for (int it = 0; it < 16; ++it) {
    const int b = 2 * it + half;
    const float gx = gridLds[wave][px][2 * b];
    const float gy = gridLds[wave][px][2 * b + 1];

    const float x = (gx + 1.0f) * (kW * 0.5f) - 0.5f;
    const float y = (gy + 1.0f) * (kH * 0.5f) - 0.5f;
    const float x0f = floorf(x), y0f = floorf(y);
    const float fx = x - x0f, fy = y - y0f;
    const int x0 = (int)x0f, y0 = (int)y0f;
    const int x1 = x0 + 1,  y1 = y0 + 1;

    const float vx0 = (x0 >= 0 && x0 < kW) ? 1.0f : 0.0f;
    const float vx1 = (x1 >= 0 && x1 < kW) ? 1.0f : 0.0f;
    const float vy0 = (y0 >= 0 && y0 < kH) ? 1.0f : 0.0f;
    const float vy1 = (y1 >= 0 && y1 < kH) ? 1.0f : 0.0f;

    const int cx0 = min(max(x0, 0), kW - 1);
    const int cx1 = min(max(x1, 0), kW - 1);
    const int cy0 = min(max(y0, 0), kH - 1);
    const int cy1 = min(max(y1, 0), kH - 1);

    const float m00 = (1.0f - fx) * (1.0f - fy) * vx0 * vy0;
    const float m01 = fx * (1.0f - fy) * vx1 * vy0;
    const float m10 = (1.0f - fx) * fy * vx0 * vy1;
    const float m11 = fx * fy * vx1 * vy1;
    const float mask = m00 + m01 + m10 + m11;
    const float bg = kCanvas * (1.0f - mask);

    const int i00 = cy0 * kW + cx0;
    const int i01 = cy0 * kW + cx1;
    const int i10 = cy1 * kW + cx0;
    const int i11 = cy1 * kW + cx1;

#pragma unroll
    for (int c = 0; c < kC; ++c) {
      const float* __restrict__ plane = inputs + (size_t)(b * kC + c) * kHW;
      const float s = plane[i00] * m00 + plane[i01] * m01 +
                      plane[i10] * m10 + plane[i11] * m11;
      out[(size_t)(b * kC + c) * kHW + p] = s * mask + bg;   // reference multiplies by mask again
    }
  }
}

extern "C" void kernel_launch(void* const* d_in, const int* in_sizes, int n_in,
                              void* d_out, int out_size, void* d_ws, size_t ws_size,
                              hipStream_t stream) {
  const float* inputs = (const float*)d_in[0];  // (32,3,512,512) f32
  const float* src    = (const float*)d_in[1];  // (32,16,2) f32
  const float* mk     = (const float*)d_in[2];  // (262144,19) f32
  float* out          = (float*)d_out;          // (32,3,512,512) f32

  const int numTiles = kHW / 16;                // 16384 waves
  dim3 grid(numTiles / 8);                      // 8 waves (256 threads) per block, exact cover
  tps_wmma_sample_kernel<<<grid, 256, 0, stream>>>(inputs, src, mk, out);
}